// GroupVectorQuantizer_16063177687196
// MI455X (gfx1250) — compile-verified
//
#include <hip/hip_runtime.h>
#include <hip/hip_bf16.h>

// ---- static problem config (from reference) ----
#define NTOK   16384
#define EDIM   256
#define NCODE  5120
#define SHARE  2048
#define BIGV   1.0e7f

typedef __attribute__((ext_vector_type(2))) float v2f;
typedef __attribute__((ext_vector_type(8))) float v8f;
typedef __attribute__((ext_vector_type(4))) int   v4i;

#if defined(__gfx1250__) && \
    __has_builtin(__builtin_amdgcn_global_load_async_to_lds_b128) && \
    __has_builtin(__builtin_amdgcn_s_wait_asynccnt)
#define USE_ASYNC_LDS 1
typedef __attribute__((address_space(1))) v4i gv4i;   // global int4
typedef __attribute__((address_space(3))) v4i lv4i;   // LDS int4
#else
#define USE_ASYNC_LDS 0
#endif

// Order-preserving (value, index) packing: u64 min == (min value, then min col).
__device__ __forceinline__ unsigned long long packdi(float v, int c) {
  unsigned int b = __float_as_uint(v);
  b = (b & 0x80000000u) ? ~b : (b | 0x80000000u);
  return ((unsigned long long)b << 32) | (unsigned int)c;
}

// ---------------------------------------------------------------------------
// Kernel 1: row norms of x and emb; init pmin to +inf; zero loss accumulators.
// One wave (32 lanes) per row; 8 rows per 256-thread block.
// ---------------------------------------------------------------------------
__global__ __launch_bounds__(256) void norms_kernel(
    const float* __restrict__ x, const float* __restrict__ emb,
    float* __restrict__ xx, float* __restrict__ ee,
    unsigned long long* __restrict__ pmin, float* __restrict__ lacc) {
  if (blockIdx.x == 0 && threadIdx.x < 4) lacc[threadIdx.x] = 0.0f;
  const int g = blockIdx.x * 256 + threadIdx.x;
  if (g < NTOK) pmin[g] = 0xFFFFFFFFFFFFFFFFull;

  const int wid  = threadIdx.x >> 5;
  const int lane = threadIdx.x & 31;
  const int r = blockIdx.x * 8 + wid;           // 0 .. 21503
  const float* src;
  float* dst;
  int row;
  if (r < NTOK) { src = x;   row = r;         dst = xx + r; }
  else          { src = emb; row = r - NTOK;  dst = ee + (r - NTOK); }
  const float* p = src + (size_t)row * EDIM;
  float s = 0.0f;
#pragma unroll
  for (int j = 0; j < 8; ++j) {
    float v = p[lane + 32 * j];
    s += v * v;
  }
#pragma unroll
  for (int off = 16; off > 0; off >>= 1) s += __shfl_xor(s, off, 32);
  if (lane == 0) *dst = s;
}

// ---------------------------------------------------------------------------
// Kernel 2: distance matrix via fp32 WMMA (16x16x4) + fused row-argmin.
// Block tile 128x128, BK=32. 8 waves in a 2(M) x 4(N) grid,
// each wave computes 64x32 = 4x2 subtiles of 16x16.
// ---------------------------------------------------------------------------
#define BM  128
#define BN  128
#define BK  32
#define LDT 36   // padded LDS row stride (floats); 36*4=144 bytes, 16B aligned

__global__ __launch_bounds__(256) void gemm_dist_kernel(
    const float* __restrict__ x, const float* __restrict__ emb,
    const long long* __restrict__ split,
    const float* __restrict__ xx, const float* __restrict__ ee,
    float* __restrict__ out_d, unsigned long long* __restrict__ pmin) {
  __shared__ __align__(16) float As[BM * LDT];
  __shared__ __align__(16) float Bs[BN * LDT];

  const int tid  = threadIdx.x;
  const int lane = tid & 31;
  const int wid  = tid >> 5;
  const int wm   = wid >> 2;            // 0..1
  const int wn   = wid & 3;             // 0..3
  const int m0   = blockIdx.x * BM;     // token tile base
  const int n0   = blockIdx.y * BN;     // code tile base
  const int hl   = lane & 15;           // lane within half-wave
  const int kho  = (lane >> 4) << 1;    // 0 or 2: K sub-offset per ISA layout

  v8f acc[4][2];
#pragma unroll
  for (int mi = 0; mi < 4; ++mi)
#pragma unroll
    for (int ni = 0; ni < 2; ++ni)
#pragma unroll
      for (int r = 0; r < 8; ++r) acc[mi][ni][r] = 0.0f;

  for (int kt = 0; kt < EDIM; kt += BK) {
    // ---- stage A (x rows) and B (emb rows) tiles into LDS: 128x32 each ----
#pragma unroll
    for (int i = 0; i < 4; ++i) {
      const int lin = tid + i * 256;        // 0..1023
      const int row = lin >> 3;             // 0..127
      const int c4  = (lin & 7) << 2;       // 0,4,..,28
      const float* gA = x   + (size_t)(m0 + row) * EDIM + kt + c4;
      const float* gB = emb + (size_t)(n0 + row) * EDIM + kt + c4;
#if USE_ASYNC_LDS
      __builtin_amdgcn_global_load_async_to_lds_b128(
          (gv4i*)gA, (lv4i*)(As + row * LDT + c4), 0, 0);
      __builtin_amdgcn_global_load_async_to_lds_b128(
          (gv4i*)gB, (lv4i*)(Bs + row * LDT + c4), 0, 0);
#else
      const float4 va = *(const float4*)gA;
      const float4 vb = *(const float4*)gB;
      *(float4*)(As + row * LDT + c4) = va;
      *(float4*)(Bs + row * LDT + c4) = vb;
#endif
      if (kt + BK < EDIM) {   // prefetch next K tile -> global_prefetch_b8
        __builtin_prefetch(gA + BK, 0, 3);
        __builtin_prefetch(gB + BK, 0, 3);
      }
    }
#if USE_ASYNC_LDS
    __builtin_amdgcn_s_wait_asynccnt(0);
#endif
    __syncthreads();

    // ---- 8 K-steps of 4, 8 WMMAs each ----
#pragma unroll
    for (int k0 = 0; k0 < BK; k0 += 4) {
      const int kk = k0 + kho;
      v2f a[4], b[2];
#pragma unroll
      for (int mi = 0; mi < 4; ++mi)
        a[mi] = *(const v2f*)(As + (wm * 64 + mi * 16 + hl) * LDT + kk);
#pragma unroll
      for (int ni = 0; ni < 2; ++ni)
        b[ni] = *(const v2f*)(Bs + (wn * 32 + ni * 16 + hl) * LDT + kk);
#pragma unroll
      for (int mi = 0; mi < 4; ++mi)
#pragma unroll
        for (int ni = 0; ni < 2; ++ni)
          acc[mi][ni] = __builtin_amdgcn_wmma_f32_16x16x4_f32(
              false, a[mi], false, b[ni], (short)0, acc[mi][ni], false, false);
    }
    __syncthreads();
  }

  // ---- epilogue: d = xx + ee - 2*dot, modality mask, NT-store, fused argmin ----
  const int s1 = (int)split[1];
  const int s2 = (int)split[2];
  int   gcolA[2];
  int   cmA[2];
  float enA[2];
#pragma unroll
  for (int ni = 0; ni < 2; ++ni) {
    gcolA[ni] = n0 + wn * 32 + ni * 16 + hl;
    cmA[ni]   = (gcolA[ni] < SHARE) ? -1 : ((gcolA[ni] - SHARE) >> 10);
    enA[ni]   = ee[gcolA[ni]];
  }
  const int rowoff = m0 + wm * 64 + ((lane >> 4) << 3);
#pragma unroll
  for (int mi = 0; mi < 4; ++mi) {
#pragma unroll
    for (int r = 0; r < 8; ++r) {
      const int grow = rowoff + mi * 16 + r;
      const int tm   = (grow >= s1) + (grow >= s2);
      const float xr = xx[grow];
      float dv0 = xr + enA[0] - 2.0f * acc[mi][0][r];
      float dv1 = xr + enA[1] - 2.0f * acc[mi][1][r];
      if (!(cmA[0] < 0 || cmA[0] == tm)) dv0 = BIGV;
      if (!(cmA[1] < 0 || cmA[1] == tm)) dv1 = BIGV;
      // d is write-only from here on: stream it past the caches
      __builtin_nontemporal_store(dv0, out_d + (size_t)grow * NCODE + gcolA[0]);
      __builtin_nontemporal_store(dv1, out_d + (size_t)grow * NCODE + gcolA[1]);
      // wave-local argmin over this row's 32 columns
      float bv = dv0;
      int   bc = gcolA[0];
      if (dv1 < bv) { bv = dv1; bc = gcolA[1]; }   // col0 < col1: keep first on tie
#pragma unroll
      for (int off = 8; off > 0; off >>= 1) {      // stays within 16-lane half
        const float ov = __shfl_xor(bv, off, 32);
        const int   oc = __shfl_xor(bc, off, 32);
        if (ov < bv || (ov == bv && oc < bc)) { bv = ov; bc = oc; }
      }
      if (hl == 0) atomicMin(pmin + grow, packdi(bv, bc));
    }
  }
}

// ---------------------------------------------------------------------------
// Kernel 3: wave per token: read packed argmin, gather code row, STE output,
// accumulate per-modality squared error.
// ---------------------------------------------------------------------------
__global__ __launch_bounds__(256) void gather_kernel(
    const float* __restrict__ x, const float* __restrict__ emb,
    const unsigned long long* __restrict__ pmin,
    const long long* __restrict__ split,
    float* __restrict__ out_xq, float* __restrict__ out_idx,
    float* __restrict__ lacc) {
  const int wid  = threadIdx.x >> 5;
  const int lane = threadIdx.x & 31;
  const int t = blockIdx.x * 8 + wid;           // 0..16383
  const int idx = (int)(pmin[t] & 0xFFFFFFFFull);
  const float* xp = x   + (size_t)t   * EDIM;
  const float* qp = emb + (size_t)idx * EDIM;
  float se = 0.0f;
#pragma unroll
  for (int j = 0; j < 8; ++j) {
    const int c = lane + 32 * j;
    const float xe = xp[c];
    const float q  = qp[c];
    out_xq[(size_t)t * EDIM + c] = xe + (q - xe);   // straight-through value
    const float d = q - xe;
    se += d * d;
  }
#pragma unroll
  for (int off = 16; off > 0; off >>= 1) se += __shfl_xor(se, off, 32);
  if (lane == 0) {
    out_idx[t] = (float)idx;
    const int tm = (t >= (int)split[1]) + (t >= (int)split[2]);
    atomicAdd(&lacc[tm], se);
  }
}

// ---------------------------------------------------------------------------
// Kernel 4: q_losses[m] = seg/cnt + 0.25*(seg/cnt)
// ---------------------------------------------------------------------------
__global__ void finalize_kernel(const long long* __restrict__ split,
                                const float* __restrict__ lacc,
                                float* __restrict__ out_q) {
  const int m = threadIdx.x;
  if (m < 3) {
    const float cnt = (float)(split[m + 1] - split[m]) * (float)EDIM;
    const float a = lacc[m] / cnt;
    out_q[m] = a + 0.25f * a;
  }
}

// ---------------------------------------------------------------------------
extern "C" void kernel_launch(void* const* d_in, const int* in_sizes, int n_in,
                              void* d_out, int out_size, void* d_ws, size_t ws_size,
                              hipStream_t stream) {
  const float*     x     = (const float*)d_in[0];      // [16384,256]
  const float*     emb   = (const float*)d_in[1];      // [5120,256]
  const long long* split = (const long long*)d_in[2];  // [4] int64

  float* out    = (float*)d_out;
  float* out_xq = out;                                  // N*256
  float* out_ix = out_xq + (size_t)NTOK * EDIM;         // N
  float* out_dm = out_ix + NTOK;                        // N*5120
  float* out_q  = out_dm + (size_t)NTOK * NCODE;        // 3

  unsigned long long* pmin = (unsigned long long*)d_ws; // [NTOK] u64
  float* xx   = (float*)(pmin + NTOK);                  // [NTOK]
  float* ee   = xx + NTOK;                              // [NCODE]
  float* lacc = ee + NCODE;                             // [4]

  norms_kernel<<<(NTOK + NCODE) / 8, 256, 0, stream>>>(x, emb, xx, ee, pmin, lacc);

  dim3 grid(NTOK / BM, NCODE / BN);  // 128 x 40
  gemm_dist_kernel<<<grid, 256, 0, stream>>>(x, emb, split, xx, ee, out_dm, pmin);

  gather_kernel<<<NTOK / 8, 256, 0, stream>>>(x, emb, pmin, split,
                                              out_xq, out_ix, lacc);

  finalize_kernel<<<1, 32, 0, stream>>>(split, lacc, out_q);
}